// CausalSelfAttentionLoRA_36034775614008
// MI455X (gfx1250) — compile-verified
//
#include <hip/hip_runtime.h>

// ---------------------------------------------------------------------------
// Causal self-attention with LoRA (q,v targets), fp32, for gfx1250 (CDNA5).
// B=2, T=2048, DM=1024, H=16, HD=64, R=8, scaling=2.0
// Uses V_WMMA_F32_16X16X4_F32 for all matmuls (fp32-exact path).
// ---------------------------------------------------------------------------

typedef __attribute__((ext_vector_type(2))) float v2f;
typedef __attribute__((ext_vector_type(8))) float v8f;

constexpr int kDM = 1024;
constexpr int kT  = 2048;
constexpr int kB  = 2;
constexpr int kH  = 16;
constexpr int kHD = 64;
constexpr int kR  = 8;
constexpr int kNT = kB * kT;          // 4096 rows
constexpr float kScaling = 2.0f;      // lora_alpha / r
constexpr float kNegInf  = -1e9f;
constexpr float kInvSqrtD = 0.125f;   // 64^-0.5

static __device__ __forceinline__ v8f wmma4(v2f a, v2f b, v8f c) {
  // D = A(16x4,f32) * B(4x16,f32) + C(16x16,f32)
  return __builtin_amdgcn_wmma_f32_16x16x4_f32(
      /*neg_a=*/false, a, /*neg_b=*/false, b,
      /*c_mod=*/(short)0, c, /*reuse_a=*/false, /*reuse_b=*/false);
}

static __device__ __forceinline__ v8f zero8() {
  v8f z = {0.f, 0.f, 0.f, 0.f, 0.f, 0.f, 0.f, 0.f};
  return z;
}

// ---------------------------------------------------------------------------
// Kernel 0: XAq = x @ Aq^T, XAv = x @ Av^T   ([4096, 8] each)
// One block (256 thr) per row; wave32 xor-reductions then cross-wave in LDS.
// ---------------------------------------------------------------------------
__global__ __launch_bounds__(256) void k_lora_xa(
    const float* __restrict__ x, const float* __restrict__ Aq,
    const float* __restrict__ Av, float* __restrict__ XAq,
    float* __restrict__ XAv) {
  const int n = blockIdx.x;
  const int tid = threadIdx.x;
  const int wave = tid >> 5, lane = tid & 31;

  float pq[kR], pv[kR];
#pragma unroll
  for (int r = 0; r < kR; ++r) { pq[r] = 0.f; pv[r] = 0.f; }

  const float* xr = x + (size_t)n * kDM;
  for (int k = tid; k < kDM; k += 256) {
    float xv = xr[k];
#pragma unroll
    for (int r = 0; r < kR; ++r) {
      pq[r] += xv * Aq[r * kDM + k];
      pv[r] += xv * Av[r * kDM + k];
    }
  }
#pragma unroll
  for (int r = 0; r < kR; ++r) {
#pragma unroll
    for (int off = 16; off > 0; off >>= 1) {
      pq[r] += __shfl_xor(pq[r], off, 32);
      pv[r] += __shfl_xor(pv[r], off, 32);
    }
  }
  __shared__ float red[8][2 * kR];
  if (lane == 0) {
#pragma unroll
    for (int r = 0; r < kR; ++r) {
      red[wave][r] = pq[r];
      red[wave][kR + r] = pv[r];
    }
  }
  __syncthreads();
  if (tid < 2 * kR) {
    float s = 0.f;
#pragma unroll
    for (int w = 0; w < 8; ++w) s += red[w][tid];
    if (tid < kR) XAq[(size_t)n * kR + tid] = s;
    else          XAv[(size_t)n * kR + (tid - kR)] = s;
  }
}

// ---------------------------------------------------------------------------
// Kernel 1: fused QKV projection with WMMA f32 16x16x4.
//   out = x @ W^T + bias (+ scaling * XA @ Bl^T for q and v)
// Block = 256 thr (8 waves), computes a 64x64 output tile; z selects q/k/v.
// LDS staging with +2 float padding to avoid bank conflicts on b64 reads.
// ---------------------------------------------------------------------------
__global__ __launch_bounds__(256) void k_qkv(
    const float* __restrict__ x,
    const float* __restrict__ Wq, const float* __restrict__ bq,
    const float* __restrict__ Wk, const float* __restrict__ bk,
    const float* __restrict__ Wv, const float* __restrict__ bv,
    const float* __restrict__ XAq, const float* __restrict__ Bq,
    const float* __restrict__ XAv, const float* __restrict__ Bv,
    float* __restrict__ qb, float* __restrict__ kb, float* __restrict__ vb) {
  const int tid = threadIdx.x;
  const int wave = tid >> 5, lane = tid & 31;
  const int row0 = blockIdx.x * 64, col0 = blockIdx.y * 64;
  const int z = blockIdx.z;

  const float* W    = (z == 0) ? Wq : (z == 1) ? Wk : Wv;
  const float* bias = (z == 0) ? bq : (z == 1) ? bk : bv;
  const float* XA   = (z == 0) ? XAq : XAv;   // unused when z==1
  const float* Bl   = (z == 0) ? Bq : Bv;     // unused when z==1
  float* outp       = (z == 0) ? qb : (z == 1) ? kb : vb;
  const bool lora = (z != 1);

  __shared__ float xs[64 * 34];   // x tile  [64 rows][32 k] padded
  __shared__ float wsm[64 * 34];  // W tile  [64 outs][32 k] padded

  // each wave owns 2 of the 16 16x16 subtiles
  const int s0 = wave, s1 = wave + 8;
  const int sr0 = s0 >> 2, sc0 = s0 & 3;
  const int sr1 = s1 >> 2, sc1 = s1 & 3;
  const int m = lane & 15, koff = (lane >> 4) * 2;

  v8f c0 = zero8(), c1 = zero8();

  for (int k0c = 0; k0c < kDM; k0c += 32) {
    __syncthreads();
    for (int i = tid; i < 64 * 32; i += 256) {
      int r = i >> 5, c = i & 31;
      xs[r * 34 + c]  = x[(size_t)(row0 + r) * kDM + k0c + c];
      wsm[r * 34 + c] = W[(size_t)(col0 + r) * kDM + k0c + c];
    }
    __syncthreads();
#pragma unroll
    for (int kk = 0; kk < 32; kk += 4) {
      v2f a0 = *(const v2f*)&xs[(sr0 * 16 + m) * 34 + kk + koff];
      v2f b0 = *(const v2f*)&wsm[(sc0 * 16 + m) * 34 + kk + koff];
      c0 = wmma4(a0, b0, c0);
      v2f a1 = *(const v2f*)&xs[(sr1 * 16 + m) * 34 + kk + koff];
      v2f b1 = *(const v2f*)&wsm[(sc1 * 16 + m) * 34 + kk + koff];
      c1 = wmma4(a1, b1, c1);
    }
  }

  // epilogue: bias + LoRA + store.  D layout: VGPR j = row j (lanes 0-15)
  // or row j+8 (lanes 16-31); lane&15 = column.
  const int g = lane >> 4, nn = lane & 15;
#pragma unroll
  for (int sidx = 0; sidx < 2; ++sidx) {
    const int sr = sidx ? sr1 : sr0;
    const int sc = sidx ? sc1 : sc0;
    const v8f cf = sidx ? c1 : c0;
    const int ocol = col0 + sc * 16 + nn;
    const float bbias = bias[ocol];
    float blc[kR];
    if (lora) {
#pragma unroll
      for (int r = 0; r < kR; ++r) blc[r] = Bl[(size_t)ocol * kR + r];
    }
#pragma unroll
    for (int j = 0; j < 8; ++j) {
      const int rM = row0 + sr * 16 + j + 8 * g;
      float acc = cf[j] + bbias;
      if (lora) {
        const float* xa = XA + (size_t)rM * kR;
        float d = 0.f;
#pragma unroll
        for (int r = 0; r < kR; ++r) d += xa[r] * blc[r];
        acc += kScaling * d;
      }
      outp[(size_t)rM * kDM + ocol] = acc;
    }
  }
}

// ---------------------------------------------------------------------------
// Kernel 2: flash attention (causal), fp32 WMMA.
// Block = 256 thr (8 waves) handles 128 query rows of one (b,h); each wave
// owns a 16-row strip. Loops over 64-key chunks <= causal limit.
// LDS: K tile region is reused as per-wave P staging after the S phase.
// ---------------------------------------------------------------------------
__global__ __launch_bounds__(256) void k_attn(
    const float* __restrict__ qb, const float* __restrict__ kb,
    const float* __restrict__ vb, const float* __restrict__ amask,
    float* __restrict__ out) {
  const int tid = threadIdx.x, wave = tid >> 5, lane = tid & 31;
  const int bh = blockIdx.x;            // 0..31
  const int b = bh / kH, h = bh % kH;
  const int qt = blockIdx.y;            // 0..15
  const int q0 = qt * 128;
  const int bT = b * kT;
  const int m = lane & 15, g = lane >> 4, koff = g * 2;
  const int qrow0 = q0 + wave * 16;

  __shared__ float KsPs[8 * 16 * 66];   // phase A: Ks[64][66]; phase B: P strips
  __shared__ float Vs[64 * 66];

  // Q strip -> A fragments (16 k-steps over HD=64), kept in registers.
  v2f aq[16];
  {
    const float* qrow = qb + (size_t)(bT + qrow0 + m) * kDM + h * kHD;
#pragma unroll
    for (int i = 0; i < 16; ++i) {
      aq[i].x = qrow[4 * i + koff];
      aq[i].y = qrow[4 * i + koff + 1];
    }
  }

  v8f oc[4];
#pragma unroll
  for (int c = 0; c < 4; ++c) oc[c] = zero8();
  float mrun[8], lrun[8];
#pragma unroll
  for (int j = 0; j < 8; ++j) { mrun[j] = -1e30f; lrun[j] = 0.f; }

  const int nchunks = 2 * qt + 2;       // keys up to q0+127 inclusive
  for (int kc = 0; kc < nchunks; ++kc) {
    const int k0 = kc * 64;
    __syncthreads();                    // protect P strips from reload
    for (int i = tid; i < 64 * 64; i += 256) {
      int r = i >> 6, c = i & 63;
      size_t src = (size_t)(bT + k0 + r) * kDM + h * kHD + c;
      KsPs[r * 66 + c] = kb[src];
      Vs[r * 66 + c]   = vb[src];
    }
    __syncthreads();

    // S = Q * K^T : four 16x16 column blocks per wave strip
    v8f sc[4];
#pragma unroll
    for (int c = 0; c < 4; ++c) {
      v8f acc = zero8();
#pragma unroll
      for (int i = 0; i < 16; ++i) {
        v2f bfr = *(const v2f*)&KsPs[(c * 16 + m) * 66 + 4 * i + koff];
        acc = wmma4(aq[i], bfr, acc);
      }
      sc[c] = acc;
    }
    __syncthreads();                    // all waves done reading Ks

    // scale + causal mask + attention mask
#pragma unroll
    for (int c = 0; c < 4; ++c) {
      const int col = k0 + c * 16 + m;
      const float amv = amask[bT + col];
#pragma unroll
      for (int j = 0; j < 8; ++j) {
        const int rM = qrow0 + j + 8 * g;
        float v = sc[c][j] * kInvSqrtD;
        v = (col > rM) ? kNegInf : v;
        sc[c][j] = v + amv;
      }
    }

    // online softmax (per row j; stats shared within 16-lane groups)
    float* Ps = &KsPs[wave * 16 * 66];
#pragma unroll
    for (int j = 0; j < 8; ++j) {
      float mx = fmaxf(fmaxf(sc[0][j], sc[1][j]), fmaxf(sc[2][j], sc[3][j]));
#pragma unroll
      for (int off = 1; off < 16; off <<= 1) mx = fmaxf(mx, __shfl_xor(mx, off, 32));
      const float mn = fmaxf(mrun[j], mx);
      const float f = __expf(mrun[j] - mn);
      float rs = 0.f;
#pragma unroll
      for (int c = 0; c < 4; ++c) {
        float p = __expf(sc[c][j] - mn);
        sc[c][j] = p;
        rs += p;
      }
#pragma unroll
      for (int off = 1; off < 16; off <<= 1) rs += __shfl_xor(rs, off, 32);
      lrun[j] = lrun[j] * f + rs;
      mrun[j] = mn;
#pragma unroll
      for (int c = 0; c < 4; ++c) oc[c][j] *= f;
    }

    // stage P (wave-private) so it can be read back in A-fragment layout
#pragma unroll
    for (int c = 0; c < 4; ++c) {
#pragma unroll
      for (int j = 0; j < 8; ++j) {
        Ps[(j + 8 * g) * 66 + c * 16 + m] = sc[c][j];
      }
    }

    // O += P * V
#pragma unroll
    for (int c2 = 0; c2 < 4; ++c2) {
      v8f acc = oc[c2];
#pragma unroll
      for (int kk = 0; kk < 64; kk += 4) {
        v2f afr = *(const v2f*)&Ps[m * 66 + kk + koff];
        v2f bfr;
        bfr.x = Vs[(kk + koff) * 66 + c2 * 16 + m];
        bfr.y = Vs[(kk + koff + 1) * 66 + c2 * 16 + m];
        acc = wmma4(afr, bfr, acc);
      }
      oc[c2] = acc;
    }
  }

  // finalize: divide by l, store [B,T,DM] with head interleave
#pragma unroll
  for (int j = 0; j < 8; ++j) {
    const float inv = 1.0f / lrun[j];
    const int rM = qrow0 + j + 8 * g;
    float* orow = out + (size_t)(bT + rM) * kDM + h * kHD;
#pragma unroll
    for (int c2 = 0; c2 < 4; ++c2) {
      orow[c2 * 16 + m] = oc[c2][j] * inv;
    }
  }
}

// ---------------------------------------------------------------------------
extern "C" void kernel_launch(void* const* d_in, const int* in_sizes, int n_in,
                              void* d_out, int out_size, void* d_ws, size_t ws_size,
                              hipStream_t stream) {
  const float* x   = (const float*)d_in[0];
  const float* am  = (const float*)d_in[1];
  const float* Wq  = (const float*)d_in[2];
  const float* bq  = (const float*)d_in[3];
  const float* Wk  = (const float*)d_in[4];
  const float* bk  = (const float*)d_in[5];
  const float* Wv  = (const float*)d_in[6];
  const float* bv  = (const float*)d_in[7];
  const float* Aq  = (const float*)d_in[8];
  const float* Bq  = (const float*)d_in[9];
  const float* Av  = (const float*)d_in[10];
  const float* Bv  = (const float*)d_in[11];
  float* out = (float*)d_out;

  float* ws  = (float*)d_ws;
  float* qb  = ws;
  float* kbf = qb  + (size_t)kNT * kDM;
  float* vbf = kbf + (size_t)kNT * kDM;
  float* XAq = vbf + (size_t)kNT * kDM;
  float* XAv = XAq + (size_t)kNT * kR;

  k_lora_xa<<<dim3(kNT), 256, 0, stream>>>(x, Aq, Av, XAq, XAv);
  k_qkv<<<dim3(kNT / 64, kDM / 64, 3), 256, 0, stream>>>(
      x, Wq, bq, Wk, bk, Wv, bv, XAq, Bq, XAv, Bv, qb, kbf, vbf);
  k_attn<<<dim3(kB * kH, kT / 128), 256, 0, stream>>>(qb, kbf, vbf, am, out);
}